// SaliencyGNN_60043642798825
// MI455X (gfx1250) — compile-verified
//
#include <hip/hip_runtime.h>

#define N_NODES 100000
#define N_EDGES 1600000
#define DIM     128
#define N_CONVS 6
#define APAD    132   // LDS row stride in floats (128 + 4 pad -> conflict-free ds_load_b64)

typedef __attribute__((ext_vector_type(2))) float v2f;
typedef __attribute__((ext_vector_type(8))) float v8f;
typedef unsigned int u32x4 __attribute__((ext_vector_type(4)));
typedef int          i32x4 __attribute__((ext_vector_type(4)));
typedef int          i32x8 __attribute__((ext_vector_type(8)));

// ---------------------------------------------------------------------------
// The TDM writes LDS through a descriptor, invisible to alias analysis.
// Escape the shared-memory pointer through an asm with a memory clobber so the
// compiler cannot fold the subsequent LDS reads to undef.
__device__ __forceinline__ void lds_publish(void* p) {
    asm volatile("" : : "v"(p) : "memory");
}

// ---------------------------------------------------------------------------
// Tensor Data Mover: DMA a 16-row x 128-col f32 tile (row stride 128 floats)
// from global memory into LDS at byte offset `lds_byte_off`, padding the LDS
// destination by 4 DWORDs after every 128 DWORDs (=> 132-float LDS row pitch).
// D# packing per CDNA5 ISA §8.3 (group0) / §8.4 (group1).
__device__ __forceinline__ void tdm_load_tile_16x128(unsigned lds_byte_off,
                                                     const float* gsrc) {
    unsigned long long ga = (unsigned long long)(size_t)gsrc;

    u32x4 g0;
    g0[0] = 1u;                                   // count=1, user descriptor
    g0[1] = lds_byte_off;                         // lds_addr [63:32]
    g0[2] = (unsigned)(ga & 0xFFFFFFFFu);         // global_addr [95:64]
    g0[3] = (unsigned)((ga >> 32) & 0x1FFFFFFu)   // global_addr [120:96]
          | (2u << 30);                           // type=2 ("image") [127:126]

    i32x8 g1;
    g1[0] = (int)((2u << 16)                      // data_size = 4 bytes
                | (1u << 20)                      // pad_enable
                | (6u << 22)                      // pad_interval: 128 DWORDs
                | (3u << 25));                    // pad_amount:   4 DWORDs
    g1[1] = (int)(128u << 16);                    // tensor_dim0[15:0] = 128
    g1[2] = (int)(16u  << 16);                    // tensor_dim1[15:0] = 16
    g1[3] = (int)(128u << 16);                    // tile_dim0 = 128 (contig row)
    g1[4] = 16;                                   // tile_dim1 = 16 rows
    g1[5] = 128;                                  // tensor_dim0_stride = 128
    g1[6] = 0;
    g1[7] = 0;

    i32x4 z4 = {0, 0, 0, 0};
#if defined(__clang_major__) && (__clang_major__ >= 23)
    i32x8 z8 = {0, 0, 0, 0, 0, 0, 0, 0};
    __builtin_amdgcn_tensor_load_to_lds(g0, g1, z4, z4, z8, 0);
#else
    __builtin_amdgcn_tensor_load_to_lds(g0, g1, z4, z4, 0);
#endif
}

// ---------------------------------------------------------------------------
__global__ void fill_zero4(float4* __restrict__ p, int n4) {
    int i = blockIdx.x * blockDim.x + threadIdx.x;
    if (i < n4) p[i] = make_float4(0.f, 0.f, 0.f, 0.f);
}

__global__ void deg_count(const long long* __restrict__ dst, float* __restrict__ deg) {
    int e = blockIdx.x * blockDim.x + threadIdx.x;
    if (e < N_EDGES) atomicAdd(&deg[(int)dst[e]], 1.0f);
}

__global__ void deg_invert(float* __restrict__ deg) {
    int i = blockIdx.x * blockDim.x + threadIdx.x;
    if (i < N_NODES) {
        float d = deg[i];
        deg[i] = 1.0f / fmaxf(d, 1.0f);
    }
}

// ---------------------------------------------------------------------------
// scatter-add of source features into agg: one wave32 per edge, float4/lane.
// x and agg are each 51.2 MB -> both L2-resident (192 MB L2); atomics are
// non-returning f32 adds at L2.
__global__ void scatter_add(const float* __restrict__ xin,
                            const long long* __restrict__ src,
                            const long long* __restrict__ dst,
                            float* __restrict__ agg) {
    unsigned tid = blockIdx.x * blockDim.x + threadIdx.x;
    unsigned e = tid >> 5;
    if (e >= N_EDGES) return;
    unsigned lane = tid & 31u;

    int s = (int)src[e];
    int d = (int)dst[e];

    const float4 v = *(const float4*)(xin + (size_t)s * DIM + lane * 4u);
    float* o = agg + (size_t)d * DIM + lane * 4u;
    atomicAdd(o + 0, v.x);
    atomicAdd(o + 1, v.y);
    atomicAdd(o + 2, v.z);
    atomicAdd(o + 3, v.w);
}

// ---------------------------------------------------------------------------
// fused SAGEConv dense part: xout = relu( (agg*deg_inv) @ Wl + xin @ Wr + b )
// One block (8 waves) per 16-node row tile; TDM stages the shared A tiles
// (agg rows, x rows) into LDS once per block; each wave owns one 16-col tile
// and runs K=128 with fp32 WMMA 16x16x4 (64 WMMAs / wave).
__global__ __launch_bounds__(256)
void sage_gemm_bias_relu(const float* __restrict__ agg,
                         const float* __restrict__ deg_inv,
                         const float* __restrict__ xin,
                         const float* __restrict__ Wl,
                         const float* __restrict__ Wr,
                         const float* __restrict__ bias,
                         float* __restrict__ xout) {
    __shared__ float smem[2 * 16 * APAD];         // [agg tile | x tile]

    const int row_tile = blockIdx.x;              // 6250 row tiles of 16 nodes
    const int wv   = threadIdx.x >> 5;            // col tile 0..7
    const int lane = threadIdx.x & 31;
    const int lo   = lane & 15;
    const int hi   = lane >> 4;                   // selects K pair / M half
    const int row_base = row_tile * 16;

    if (threadIdx.x == 0) {                       // TDM issue ignores EXEC; 1 issue/block
        tdm_load_tile_16x128(0u, agg + (size_t)row_base * DIM);
        tdm_load_tile_16x128((unsigned)(16 * APAD * sizeof(float)),
                             xin + (size_t)row_base * DIM);
    }
    __builtin_amdgcn_s_wait_tensorcnt((short)0);  // wave0: DMA landed; others: no-op
    __syncthreads();
    lds_publish(smem);                            // smem now "written" for the compiler

    const int col = wv * 16 + lo;
    const float dinv = deg_inv[row_base + lo];
    const float* __restrict__ As = smem + lo * APAD;              // agg row (lane's M)
    const float* __restrict__ Xs = smem + 16 * APAD + lo * APAD;  // x row

    v8f acc = {};

    #pragma unroll 8
    for (int k = 0; k < DIM; k += 4) {
        const int ka = k + hi * 2;                // this lane holds K = ka, ka+1

        // A fragments from LDS (conflict-free: 132-float pitch)
        v2f aL = *(const v2f*)(As + ka);
        aL.x *= dinv;                             // fold mean = agg * deg_inv
        aL.y *= dinv;
        v2f aR = *(const v2f*)(Xs + ka);

        // B fragments (4x16) from global; W matrices are L0/L2 resident
        v2f bL;
        bL.x = Wl[(size_t)ka * DIM + col];
        bL.y = Wl[(size_t)(ka + 1) * DIM + col];
        v2f bR;
        bR.x = Wr[(size_t)ka * DIM + col];
        bR.y = Wr[(size_t)(ka + 1) * DIM + col];

        acc = __builtin_amdgcn_wmma_f32_16x16x4_f32(false, aL, false, bL,
                                                    (short)0, acc, false, false);
        acc = __builtin_amdgcn_wmma_f32_16x16x4_f32(false, aR, false, bR,
                                                    (short)0, acc, false, false);
    }

    // epilogue: bias + relu; VGPR i -> M = i + hi*8, N = lo
    const float bb = bias[col];
    #pragma unroll
    for (int i = 0; i < 8; ++i) {
        float v = acc[i] + bb;
        xout[(size_t)(row_base + hi * 8 + i) * DIM + col] = v > 0.f ? v : 0.f;
    }
}

// ---------------------------------------------------------------------------
// final projection: out[n] = dot(x[n,:], W_lin) + b_lin ; one wave per node
__global__ void final_linear(const float* __restrict__ xin,
                             const float* __restrict__ wlin,
                             const float* __restrict__ blin,
                             float* __restrict__ out) {
    unsigned tid = blockIdx.x * blockDim.x + threadIdx.x;
    unsigned node = tid >> 5;
    if (node >= N_NODES) return;
    unsigned lane = tid & 31u;

    const float4 xv = *(const float4*)(xin + (size_t)node * DIM + lane * 4u);
    const float4 wv = *(const float4*)(wlin + lane * 4u);
    float s = xv.x * wv.x + xv.y * wv.y + xv.z * wv.z + xv.w * wv.w;

    #pragma unroll
    for (int off = 16; off > 0; off >>= 1)
        s += __shfl_xor(s, off, 32);

    if (lane == 0) out[node] = s + blin[0];
}

// ---------------------------------------------------------------------------
extern "C" void kernel_launch(void* const* d_in, const int* in_sizes, int n_in,
                              void* d_out, int out_size, void* d_ws, size_t ws_size,
                              hipStream_t stream) {
    (void)in_sizes; (void)n_in; (void)out_size; (void)ws_size;

    const float*     x    = (const float*)d_in[0];
    const long long* edge = (const long long*)d_in[1];   // int64 [2, N_EDGES]
    const float*     Wl   = (const float*)d_in[2];       // [6,128,128]
    const float*     Wr   = (const float*)d_in[3];       // [6,128,128]
    const float*     b    = (const float*)d_in[4];       // [6,128]
    const float*     wlin = (const float*)d_in[5];       // [128,1]
    const float*     blin = (const float*)d_in[6];       // [1]
    float*           out  = (float*)d_out;               // [100000]

    const long long* src = edge;
    const long long* dst = edge + N_EDGES;

    float* ws  = (float*)d_ws;
    float* xA  = ws;                                     // [N, D]
    float* xB  = ws + (size_t)N_NODES * DIM;             // [N, D]
    float* agg = ws + 2 * (size_t)N_NODES * DIM;         // [N, D]
    float* deg = ws + 3 * (size_t)N_NODES * DIM;         // [N] (becomes deg_inv)

    // ---- degrees -> deg_inv
    fill_zero4<<<(N_NODES / 4 + 255) / 256, 256, 0, stream>>>((float4*)deg, N_NODES / 4);
    deg_count <<<(N_EDGES + 255) / 256, 256, 0, stream>>>(dst, deg);
    deg_invert<<<(N_NODES + 255) / 256, 256, 0, stream>>>(deg);

    const int nAgg4    = (N_NODES * DIM) / 4;
    const int gridFill = (nAgg4 + 255) / 256;
    const int gridScat = (N_EDGES * 32) / 256;           // exact: 200000 blocks
    const int gridGemm = N_NODES / 16;                   // 6250 row-tile blocks

    const float* cur = x;       // layer 0 reads input directly
    float*       nxt = xA;
    for (int i = 0; i < N_CONVS; ++i) {
        fill_zero4<<<gridFill, 256, 0, stream>>>((float4*)agg, nAgg4);
        scatter_add<<<gridScat, 256, 0, stream>>>(cur, src, dst, agg);
        sage_gemm_bias_relu<<<gridGemm, 256, 0, stream>>>(
            agg, deg, cur,
            Wl + (size_t)i * DIM * DIM,
            Wr + (size_t)i * DIM * DIM,
            b  + (size_t)i * DIM,
            nxt);
        cur = nxt;
        nxt = (cur == xA) ? xB : xA;
    }

    final_linear<<<(N_NODES * 32) / 256 + 1, 256, 0, stream>>>(cur, wlin, blin, out);
}